// GroverFilteredAttention_71184787964430
// MI455X (gfx1250) — compile-verified
//
#include <hip/hip_runtime.h>
#include <hip/hip_bf16.h>

// ---------------------------------------------------------------------------
// MI455X (gfx1250) fused Grover-filtered attention, fp32 WMMA path.
// Problem: B=4, S=1024, E=1024, H=16, D=64.  Memory-bound (272MB output), so
// fp32 V_WMMA_F32_16X16X4_F32 is the right matrix op; the 320KB LDS per WGP
// lets the whole 64x1024 logits block stay on-chip through both softmax
// passes, the threshold filter, and the attn@V GEMM.
// ---------------------------------------------------------------------------

typedef __attribute__((ext_vector_type(2))) float v2f;
typedef __attribute__((ext_vector_type(8))) float v8f;

__device__ __forceinline__ v8f wmma4(v2f a, v2f b, v8f c) {
    // D = A(16x4) * B(4x16) + C, fp32, wave32
    return __builtin_amdgcn_wmma_f32_16x16x4_f32(
        /*neg_a=*/false, a, /*neg_b=*/false, b,
        /*c_mod=*/(short)0, c, /*reuse_a=*/false, /*reuse_b=*/false);
}

constexpr int Bc = 4;
constexpr int Sc = 1024;
constexpr int Ec = 1024;
constexpr int Hc = 16;
constexpr int Dc = 64;
constexpr int N3 = 3 * Ec;        // 3072
constexpr int QB = 64;            // q rows per workgroup in attention kernel
constexpr int LSTR = 1028;        // LDS row stride (floats), offsets banks
constexpr float THRESHOLD = 4e-4f;

// ---------------------------------------------------------------------------
// Kernel 1: QKV projection.  C[m,n] = sum_e x[m,e]*Wqkv[n,e] + bqkv[n]
// M = B*S = 4096, N = 3E = 3072, K = E = 1024.
// Wqkv is [N,K] row-major -> both A and B fragments are contiguous float2.
// Epilogue scatters into q/k/v laid out [B,H,S,D].
// ---------------------------------------------------------------------------
__global__ __launch_bounds__(256) void qkv_kernel(
    const float* __restrict__ x, const float* __restrict__ Wqkv,
    const float* __restrict__ bqkv,
    float* __restrict__ qd, float* __restrict__ kd, float* __restrict__ vd)
{
    const int lane = threadIdx.x & 31;
    const int wave = threadIdx.x >> 5;
    const int half = lane >> 4;      // 0: K pair {0,1}, 1: K pair {2,3}
    const int l16  = lane & 15;

    // 8 waves: 4 along M x 2 along N; each wave does a 32x32 tile.
    const int wm = wave & 3;
    const int wn = wave >> 2;
    const int m0 = blockIdx.y * 128 + wm * 32;
    const int n0 = blockIdx.x * 64  + wn * 32;

    const float* A0 = x    + (size_t)(m0 +      l16) * Ec + 2 * half;
    const float* A1 = x    + (size_t)(m0 + 16 + l16) * Ec + 2 * half;
    const float* B0 = Wqkv + (size_t)(n0 +      l16) * Ec + 2 * half;
    const float* B1 = Wqkv + (size_t)(n0 + 16 + l16) * Ec + 2 * half;

    v8f acc[2][2] = {};
#pragma unroll 4
    for (int kk = 0; kk < Ec; kk += 4) {
        v2f a0 = *(const v2f*)(A0 + kk);
        v2f a1 = *(const v2f*)(A1 + kk);
        v2f b0 = *(const v2f*)(B0 + kk);
        v2f b1 = *(const v2f*)(B1 + kk);
        acc[0][0] = wmma4(a0, b0, acc[0][0]);
        acc[0][1] = wmma4(a0, b1, acc[0][1]);
        acc[1][0] = wmma4(a1, b0, acc[1][0]);
        acc[1][1] = wmma4(a1, b1, acc[1][1]);
    }

#pragma unroll
    for (int ti = 0; ti < 2; ++ti)
#pragma unroll
    for (int tj = 0; tj < 2; ++tj) {
        const int ncol = n0 + tj * 16 + l16;
        const float bias = bqkv[ncol];
        const int t = ncol >> 10;        // 0=q 1=k 2=v   (E = 1024)
        const int r = ncol & 1023;
        const int h = r >> 6;            // D = 64
        const int d = r & 63;
        float* dst = (t == 0) ? qd : ((t == 1) ? kd : vd);
#pragma unroll
        for (int i = 0; i < 8; ++i) {
            const int m = m0 + ti * 16 + half * 8 + i;
            const int b = m >> 10;       // S = 1024
            const int s = m & 1023;
            dst[(((size_t)b * Hc + h) * Sc + s) * Dc + d] = acc[ti][tj][i] + bias;
        }
    }
}

// ---------------------------------------------------------------------------
// Kernel 2: fused attention per (b*H+h, 64-row q block).
//   phase 1: logits(64x1024) = scale * Q_blk @ K^T  -> LDS (WMMA)
//   phase 2: row max, Z, threshold T*Z, Z2 (all in LDS)
//   phase 3: filtered_attn written once to HBM, normalized copy kept in LDS
//   phase 4: ctx(64x64) = attn @ V                  (WMMA, A from LDS)
// ---------------------------------------------------------------------------
__global__ __launch_bounds__(256) void attn_kernel(
    const float* __restrict__ qd, const float* __restrict__ kd,
    const float* __restrict__ vd,
    float* __restrict__ attn_out,   // [B,H,S,S] region of d_out
    float* __restrict__ ctx)        // [B,S,E]
{
    extern __shared__ float smem[];
    float* lds    = smem;                    // QB x LSTR logits / probs
    float* rowThr = smem + QB * LSTR;        // QB: exp-domain threshold (T*Z)
    float* rowInv = rowThr + QB;             // QB: 1/Z2

    const int lane = threadIdx.x & 31;
    const int wave = threadIdx.x >> 5;
    const int half = lane >> 4;
    const int l16  = lane & 15;

    const int bh = blockIdx.y;               // b*H + h
    const int s0 = blockIdx.x * QB;

    const float* Q = qd + (size_t)bh * Sc * Dc;
    const float* K = kd + (size_t)bh * Sc * Dc;
    const float* V = vd + (size_t)bh * Sc * Dc;
    const float scale = 0.125f;              // 1/sqrt(64)

    // ---- phase 1: logits into LDS.  waves: 2(M) x 4(N); N loop stride 128.
    {
        const int wm = wave & 1;
        const int wn = wave >> 1;
        const int m0 = wm * 32;
        const float* A0 = Q + (size_t)(s0 + m0 +      l16) * Dc + 2 * half;
        const float* A1 = Q + (size_t)(s0 + m0 + 16 + l16) * Dc + 2 * half;
        for (int nblk = 0; nblk < Sc; nblk += 128) {
            const int n0 = nblk + wn * 32;
            const float* B0 = K + (size_t)(n0 +      l16) * Dc + 2 * half;
            const float* B1 = K + (size_t)(n0 + 16 + l16) * Dc + 2 * half;
            v8f acc[2][2] = {};
#pragma unroll
            for (int kk = 0; kk < Dc; kk += 4) {
                v2f a0 = *(const v2f*)(A0 + kk);
                v2f a1 = *(const v2f*)(A1 + kk);
                v2f b0 = *(const v2f*)(B0 + kk);
                v2f b1 = *(const v2f*)(B1 + kk);
                acc[0][0] = wmma4(a0, b0, acc[0][0]);
                acc[0][1] = wmma4(a0, b1, acc[0][1]);
                acc[1][0] = wmma4(a1, b0, acc[1][0]);
                acc[1][1] = wmma4(a1, b1, acc[1][1]);
            }
#pragma unroll
            for (int ti = 0; ti < 2; ++ti)
#pragma unroll
            for (int tj = 0; tj < 2; ++tj)
#pragma unroll
            for (int i = 0; i < 8; ++i) {
                const int m = m0 + ti * 16 + half * 8 + i;
                const int n = n0 + tj * 16 + l16;
                lds[m * LSTR + n] = acc[ti][tj][i] * scale;
            }
        }
    }
    __syncthreads();

    // ---- phase 2: softmax stats, 4 threads per row (same wave -> shfl).
    {
        const int r   = threadIdx.x >> 2;
        const int sub = threadIdx.x & 3;
        float* row = lds + r * LSTR;
        float mx = -3.0e38f;
        for (int c = sub; c < Sc; c += 4) mx = fmaxf(mx, row[c]);
        mx = fmaxf(mx, __shfl_xor(mx, 1, 32));
        mx = fmaxf(mx, __shfl_xor(mx, 2, 32));
        float z = 0.f;
        for (int c = sub; c < Sc; c += 4) {
            const float e = __expf(row[c] - mx);
            row[c] = e;                  // keep exp in LDS
            z += e;
        }
        z += __shfl_xor(z, 1, 32);
        z += __shfl_xor(z, 2, 32);
        const float thr = THRESHOLD * z; // p>T  <=>  e > T*Z
        float z2 = 0.f;
        for (int c = sub; c < Sc; c += 4) {
            const float e = row[c];
            if (e > thr) z2 += e;
        }
        z2 += __shfl_xor(z2, 1, 32);
        z2 += __shfl_xor(z2, 2, 32);
        if (sub == 0) { rowThr[r] = thr; rowInv[r] = 1.f / z2; }
    }
    __syncthreads();

    // ---- phase 3: single coalesced HBM write of filtered_attn; renorm LDS.
    {
        float* dstA = attn_out + ((size_t)bh * Sc + s0) * Sc;
        for (int idx = threadIdx.x; idx < QB * Sc; idx += 256) {
            const int r = idx >> 10;
            const int c = idx & (Sc - 1);
            const float e = lds[r * LSTR + c];
            const float val = (e > rowThr[r]) ? e * rowInv[r] : 0.f;
            lds[r * LSTR + c]      = val;
            dstA[(size_t)r * Sc + c] = val;
        }
    }
    __syncthreads();

    // ---- phase 4: ctx_blk[64,64] = attn(LDS) @ V.  waves 0..3: 2(M)x2(N).
    if (wave < 4) {
        const int m0 = (wave & 1) * 32;
        const int n0 = (wave >> 1) * 32;
        v8f acc[2][2] = {};
#pragma unroll 4
        for (int kk = 0; kk < Sc; kk += 4) {
            const int kb = kk + 2 * half;
            v2f a0 = *(const v2f*)(lds + (m0 +      l16) * LSTR + kb);
            v2f a1 = *(const v2f*)(lds + (m0 + 16 + l16) * LSTR + kb);
            v2f b0 = { V[(size_t)kb * Dc + n0 +      l16],
                       V[(size_t)(kb + 1) * Dc + n0 +      l16] };
            v2f b1 = { V[(size_t)kb * Dc + n0 + 16 + l16],
                       V[(size_t)(kb + 1) * Dc + n0 + 16 + l16] };
            acc[0][0] = wmma4(a0, b0, acc[0][0]);
            acc[0][1] = wmma4(a0, b1, acc[0][1]);
            acc[1][0] = wmma4(a1, b0, acc[1][0]);
            acc[1][1] = wmma4(a1, b1, acc[1][1]);
        }
        const int b = bh >> 4;           // /H
        const int h = bh & 15;
#pragma unroll
        for (int ti = 0; ti < 2; ++ti)
#pragma unroll
        for (int tj = 0; tj < 2; ++tj)
#pragma unroll
        for (int i = 0; i < 8; ++i) {
            const int m = m0 + ti * 16 + half * 8 + i;   // row in block
            const int n = n0 + tj * 16 + l16;            // d
            ctx[((size_t)b * Sc + (s0 + m)) * Ec + h * Dc + n] = acc[ti][tj][i];
        }
    }
}

// ---------------------------------------------------------------------------
// Kernel 3: output projection.  out[m,n] = sum_k ctx[m,k]*Wproj[n,k] + bproj[n]
// M = 4096, N = 1024, K = 1024.
// ---------------------------------------------------------------------------
__global__ __launch_bounds__(256) void proj_kernel(
    const float* __restrict__ ctx, const float* __restrict__ Wproj,
    const float* __restrict__ bproj, float* __restrict__ out)
{
    const int lane = threadIdx.x & 31;
    const int wave = threadIdx.x >> 5;
    const int half = lane >> 4;
    const int l16  = lane & 15;

    const int wm = wave & 3;
    const int wn = wave >> 2;
    const int m0 = blockIdx.y * 128 + wm * 32;
    const int n0 = blockIdx.x * 64  + wn * 32;

    const float* A0 = ctx   + (size_t)(m0 +      l16) * Ec + 2 * half;
    const float* A1 = ctx   + (size_t)(m0 + 16 + l16) * Ec + 2 * half;
    const float* B0 = Wproj + (size_t)(n0 +      l16) * Ec + 2 * half;
    const float* B1 = Wproj + (size_t)(n0 + 16 + l16) * Ec + 2 * half;

    v8f acc[2][2] = {};
#pragma unroll 4
    for (int kk = 0; kk < Ec; kk += 4) {
        v2f a0 = *(const v2f*)(A0 + kk);
        v2f a1 = *(const v2f*)(A1 + kk);
        v2f b0 = *(const v2f*)(B0 + kk);
        v2f b1 = *(const v2f*)(B1 + kk);
        acc[0][0] = wmma4(a0, b0, acc[0][0]);
        acc[0][1] = wmma4(a0, b1, acc[0][1]);
        acc[1][0] = wmma4(a1, b0, acc[1][0]);
        acc[1][1] = wmma4(a1, b1, acc[1][1]);
    }

#pragma unroll
    for (int ti = 0; ti < 2; ++ti)
#pragma unroll
    for (int tj = 0; tj < 2; ++tj) {
        const int n = n0 + tj * 16 + l16;
        const float bias = bproj[n];
#pragma unroll
        for (int i = 0; i < 8; ++i) {
            const int m = m0 + ti * 16 + half * 8 + i;
            out[(size_t)m * Ec + n] = acc[ti][tj][i] + bias;
        }
    }
}

// ---------------------------------------------------------------------------
extern "C" void kernel_launch(void* const* d_in, const int* in_sizes, int n_in,
                              void* d_out, int out_size, void* d_ws, size_t ws_size,
                              hipStream_t stream) {
    (void)in_sizes; (void)n_in; (void)out_size; (void)ws_size;

    const float* x     = (const float*)d_in[0];
    const float* Wqkv  = (const float*)d_in[1];
    const float* bqkv  = (const float*)d_in[2];
    const float* Wproj = (const float*)d_in[3];
    const float* bproj = (const float*)d_in[4];

    float* out  = (float*)d_out;                          // [B,S,E]
    float* attn = out + (size_t)Bc * Sc * Ec;             // [B,H,S,S]

    const size_t headElems = (size_t)Bc * Hc * Sc * Dc;   // 4M floats
    float* qd  = (float*)d_ws;
    float* kd  = qd + headElems;
    float* vd  = kd + headElems;
    float* ctx = vd + headElems;                          // [B,S,E]

    const size_t smemBytes = (size_t)(QB * LSTR + 2 * QB) * sizeof(float); // ~264KB
    (void)hipFuncSetAttribute(reinterpret_cast<const void*>(attn_kernel),
                              hipFuncAttributeMaxDynamicSharedMemorySize,
                              (int)smemBytes);

    qkv_kernel <<<dim3(N3 / 64, (Bc * Sc) / 128), 256, 0, stream>>>(x, Wqkv, bqkv, qd, kd, vd);
    attn_kernel<<<dim3(Sc / QB, Bc * Hc), 256, smemBytes, stream>>>(qd, kd, vd, attn, ctx);
    proj_kernel<<<dim3(Ec / 64, (Bc * Sc) / 128), 256, 0, stream>>>(ctx, Wproj, bproj, out);
}